// GCNLayer_52304111730900
// MI455X (gfx1250) — compile-verified
//
#include <hip/hip_runtime.h>
#include <hip/hip_bf16.h>
#include <math.h>

// Problem constants (match reference)
constexpr int Bc = 64, Nc = 325, Tc = 24, Fc = 64, Hc = 64, Ec = 2600;
constexpr int Gc = Bc * Tc;          // 1536 folded graphs
constexpr int Rc = Gc * Nc;          // 499200 (G,N) rows
constexpr float NEG_SLOPE = 0.2f;

typedef __attribute__((ext_vector_type(2))) float v2f;
typedef __attribute__((ext_vector_type(8))) float v8f;

// ---------------------------------------------------------------------------
// Kernel 1: h = x * W^T via V_WMMA_F32_16X16X4_F32, fused attention logits.
// Wave computes a 16-row x 64-col tile. Rows indexed r = g*N + n; the source
// row of x lives at ((b*N + n)*T + t)*F with g = b*T + t (transpose fold).
// A layout (f32 16x4, ISA 7.12.2): lanes 0-15 = rows M, K pair {0,1};
// lanes 16-31 = rows M, K pair {2,3}. B (4x16) mirrors the C/D half-lane
// split: VGPR j holds K = 4*kt + j + 2*(lane>=16), col = lane%16.
// ---------------------------------------------------------------------------
__global__ __launch_bounds__(256)
void gat_gemm_att(const float* __restrict__ x, const float* __restrict__ W,
                  const float* __restrict__ att_src, const float* __restrict__ att_dst,
                  float* __restrict__ h_ws, float* __restrict__ as_ws,
                  float* __restrict__ ad_ws)
{
    __shared__ v2f ldsW[64 * 32];   // [j*16 + kt][lane] B-operand pairs, 16 KB

    const int tid  = threadIdx.x;
    const int lane = tid & 31;
    const int wave = tid >> 5;

    // Stage W (64x64 f32, row-major [h][f]) into WMMA B layout.
    for (int s = tid; s < 64 * 32; s += 256) {
        const int sl = s & 31;
        const int jk = s >> 5;          // j*16 + kt
        const int j  = jk >> 4;
        const int kt = jk & 15;
        const int ncol = 16 * j + (sl & 15);
        const int k    = kt * 4 + 2 * (sl >> 4);
        v2f w2;
        w2.x = W[ncol * Fc + k];
        w2.y = W[ncol * Fc + k + 1];
        ldsW[s] = w2;
    }
    __syncthreads();

    const int tile = blockIdx.x * 8 + wave;     // 31200 tiles total
    const int r0   = tile * 16;

    // This lane's A-matrix row.
    const int m = lane & 15;
    const int r = r0 + m;
    const int g = r / Nc;
    const int n = r - g * Nc;
    const int b = g / Tc;
    const int t = g - b * Tc;
    const float* __restrict__ xrow =
        x + (((size_t)(b * Nc + n)) * Tc + t) * Fc;
    const int khalf = 2 * (lane >> 4);

    v8f c0 = {}, c1 = {}, c2 = {}, c3 = {};

#pragma unroll
    for (int kt = 0; kt < 16; ++kt) {
        const int k = kt * 4 + khalf;
        v2f a;
        a.x = xrow[k];
        a.y = xrow[k + 1];
        const v2f b0 = ldsW[(0 * 16 + kt) * 32 + lane];
        const v2f b1 = ldsW[(1 * 16 + kt) * 32 + lane];
        const v2f b2 = ldsW[(2 * 16 + kt) * 32 + lane];
        const v2f b3 = ldsW[(3 * 16 + kt) * 32 + lane];
        c0 = __builtin_amdgcn_wmma_f32_16x16x4_f32(false, a, false, b0, (short)0, c0, false, false);
        c1 = __builtin_amdgcn_wmma_f32_16x16x4_f32(false, a, false, b1, (short)0, c1, false, false);
        c2 = __builtin_amdgcn_wmma_f32_16x16x4_f32(false, a, false, b2, (short)0, c2, false, false);
        c3 = __builtin_amdgcn_wmma_f32_16x16x4_f32(false, a, false, b3, (short)0, c3, false, false);
    }

    // C/D layout: VGPR rr, lane L -> row (rr + 8*(L>=16)), col (16j + L%16).
    const int rowHalf = 8 * (lane >> 4);
    const int colL    = lane & 15;

#pragma unroll
    for (int rr = 0; rr < 8; ++rr) {
        const int rg = r0 + rr + rowHalf;
        float* hp = h_ws + (size_t)rg * Hc;
        hp[ 0 + colL] = c0[rr];
        hp[16 + colL] = c1[rr];
        hp[32 + colL] = c2[rr];
        hp[48 + colL] = c3[rr];
    }

    // Fused attention logits: a_s[r] = h[r,:]·att_src, a_d[r] = h[r,:]·att_dst.
    const float ws0 = att_src[ 0 + colL], ws1 = att_src[16 + colL];
    const float ws2 = att_src[32 + colL], ws3 = att_src[48 + colL];
    const float wd0 = att_dst[ 0 + colL], wd1 = att_dst[16 + colL];
    const float wd2 = att_dst[32 + colL], wd3 = att_dst[48 + colL];

#pragma unroll
    for (int rr = 0; rr < 8; ++rr) {
        float ps = c0[rr] * ws0 + c1[rr] * ws1 + c2[rr] * ws2 + c3[rr] * ws3;
        float pd = c0[rr] * wd0 + c1[rr] * wd1 + c2[rr] * wd2 + c3[rr] * wd3;
#pragma unroll
        for (int off = 8; off >= 1; off >>= 1) {   // reduce within 16-lane half
            ps += __shfl_xor(ps, off, 32);
            pd += __shfl_xor(pd, off, 32);
        }
        if (colL == 0) {
            const int rg = r0 + rr + rowHalf;
            as_ws[rg] = ps;
            ad_ws[rg] = pd;
        }
    }
}

// ---------------------------------------------------------------------------
// Kernel 2: init output accumulator + segment-max/denominator state.
// ---------------------------------------------------------------------------
__global__ void gat_init(float* __restrict__ out, unsigned* __restrict__ m_enc,
                         float* __restrict__ den)
{
    const int idx = blockIdx.x * 256 + threadIdx.x;
    if (idx < Rc * Hc) out[idx] = 0.0f;
    if (idx < Rc) { m_enc[idx] = 0u; den[idx] = 0.0f; }
}

__device__ __forceinline__ unsigned enc_f32(float f) {
    unsigned u = __float_as_uint(f);
    return (u & 0x80000000u) ? ~u : (u | 0x80000000u);   // monotonic order map
}
__device__ __forceinline__ float dec_f32(unsigned key) {
    unsigned u = (key & 0x80000000u) ? (key & 0x7FFFFFFFu) : ~key;
    return __uint_as_float(u);
}

// ---------------------------------------------------------------------------
// Kernel 3: per-(g,edge) leaky-relu score + segment max over dst (atomicMax).
// ---------------------------------------------------------------------------
__global__ void gat_edge_max(const float* __restrict__ as_ws, const float* __restrict__ ad_ws,
                             const int* __restrict__ edge, float* __restrict__ e_buf,
                             unsigned* __restrict__ m_enc)
{
    const int idx = blockIdx.x * 256 + threadIdx.x;
    if (idx >= Gc * Ec) return;
    const int g = idx / Ec;
    const int e = idx - g * Ec;
    const int src = edge[e];
    const int dst = edge[Ec + e];
    float sc = as_ws[g * Nc + src] + ad_ws[g * Nc + dst];
    sc = sc > 0.0f ? sc : NEG_SLOPE * sc;
    e_buf[idx] = sc;
    atomicMax(&m_enc[g * Nc + dst], enc_f32(sc));
}

// ---------------------------------------------------------------------------
// Kernel 4: ex = exp(score - max[dst]); denominators via atomicAdd.
// ---------------------------------------------------------------------------
__global__ void gat_edge_exp(const int* __restrict__ edge, float* __restrict__ e_buf,
                             const unsigned* __restrict__ m_enc, float* __restrict__ den)
{
    const int idx = blockIdx.x * 256 + threadIdx.x;
    if (idx >= Gc * Ec) return;
    const int g = idx / Ec;
    const int e = idx - g * Ec;
    const int dst = edge[Ec + e];
    const float mx = dec_f32(m_enc[g * Nc + dst]);
    const float ex = __expf(e_buf[idx] - mx);
    e_buf[idx] = ex;
    atomicAdd(&den[g * Nc + dst], ex);
}

// ---------------------------------------------------------------------------
// Kernel 5: alpha-weighted scatter of h[src] into out[dst]; 64 threads/edge,
// one per feature channel (coalesced gathers + f32 global atomics -> L2).
// ---------------------------------------------------------------------------
__global__ void gat_scatter(const int* __restrict__ edge, const float* __restrict__ e_buf,
                            const float* __restrict__ den, const float* __restrict__ h_ws,
                            float* __restrict__ out)
{
    const long long idx = (long long)blockIdx.x * 256 + threadIdx.x;
    if (idx >= (long long)Gc * Ec * Hc) return;
    const int hch  = (int)(idx & 63);
    const int pair = (int)(idx >> 6);
    const int g = pair / Ec;
    const int e = pair - g * Ec;
    const int src = edge[e];
    const int dst = edge[Ec + e];
    const float alpha = e_buf[pair] / den[g * Nc + dst];
    const float msg = alpha * h_ws[((size_t)(g * Nc + src)) * Hc + hch];
    atomicAdd(&out[((size_t)(g * Nc + dst)) * Hc + hch], msg);
}

// ---------------------------------------------------------------------------
// Kernel 6: out = gelu(acc + bias), exact erf form (approximate=False).
// ---------------------------------------------------------------------------
__global__ void gat_gelu(float* __restrict__ out, const float* __restrict__ bias)
{
    const int idx = blockIdx.x * 256 + threadIdx.x;
    if (idx >= Rc * Hc) return;
    const float v = out[idx] + bias[idx & 63];
    out[idx] = 0.5f * v * (1.0f + erff(v * 0.70710678118654752f));
}

// ---------------------------------------------------------------------------
extern "C" void kernel_launch(void* const* d_in, const int* in_sizes, int n_in,
                              void* d_out, int out_size, void* d_ws, size_t ws_size,
                              hipStream_t stream)
{
    (void)in_sizes; (void)n_in; (void)out_size; (void)ws_size;

    const float* x        = (const float*)d_in[0];
    const int*   edge     = (const int*)  d_in[1];
    const float* W        = (const float*)d_in[2];
    const float* att_src  = (const float*)d_in[3];
    const float* att_dst  = (const float*)d_in[4];
    const float* bias     = (const float*)d_in[5];
    float* out = (float*)d_out;

    // Workspace carve-up (bytes): h | a_s | a_d | m_enc | den | e_buf  (~152 MB)
    char* ws = (char*)d_ws;
    float*    h_ws  = (float*)ws;                      ws += (size_t)Rc * Hc * 4;
    float*    as_ws = (float*)ws;                      ws += (size_t)Rc * 4;
    float*    ad_ws = (float*)ws;                      ws += (size_t)Rc * 4;
    unsigned* m_enc = (unsigned*)ws;                   ws += (size_t)Rc * 4;
    float*    den   = (float*)ws;                      ws += (size_t)Rc * 4;
    float*    e_buf = (float*)ws;

    // 1) GEMM h = x W^T (WMMA f32) + fused attention logits.
    //    Rc/16 = 31200 wave-tiles, 8 waves per 256-thread block -> 3900 blocks.
    gat_gemm_att<<<3900, 256, 0, stream>>>(x, W, att_src, att_dst, h_ws, as_ws, ad_ws);

    // 2) Init accumulators.
    {
        const int total = Rc * Hc;
        gat_init<<<(total + 255) / 256, 256, 0, stream>>>(out, m_enc, den);
    }

    // 3) Edge scores + segment max.
    {
        const int total = Gc * Ec;
        gat_edge_max<<<(total + 255) / 256, 256, 0, stream>>>(as_ws, ad_ws, edge, e_buf, m_enc);
        // 4) exp + denominators.
        gat_edge_exp<<<(total + 255) / 256, 256, 0, stream>>>(edge, e_buf, m_enc, den);
    }

    // 5) Weighted message scatter (64 channels per edge).
    {
        const long long total = (long long)Gc * Ec * Hc;
        gat_scatter<<<(int)((total + 255) / 256), 256, 0, stream>>>(edge, e_buf, den, h_ws, out);
    }

    // 6) Bias + exact GELU, in place on d_out ([G,N,H] flat == reference reshape).
    {
        const int total = Rc * Hc;
        gat_gelu<<<(total + 255) / 256, 256, 0, stream>>>(out, bias);
    }
}